// GCNnet_76416058131444
// MI455X (gfx1250) — compile-verified
//
#include <hip/hip_runtime.h>
#include <hip/hip_bf16.h>

typedef float v2f __attribute__((ext_vector_type(2)));
typedef float v8f __attribute__((ext_vector_type(8)));

#define DIM     128
#define NGRAPH  64
#define ROWT    5            // row tiles (of 16) per block: 3125 = 625 * 5
#define TROWS   (ROWT * 16)  // 80 rows staged per block
#define LSTR    132          // LDS row stride (floats): conflict-free b64 reads

// ---------------------------------------------------------------------------
// Utility: zero a float buffer
// ---------------------------------------------------------------------------
__global__ __launch_bounds__(256) void zero_f32(float* __restrict__ p, int n) {
  int t = blockIdx.x * blockDim.x + threadIdx.x;
  if (t < n) p[t] = 0.0f;
}

// ---------------------------------------------------------------------------
// Degree: deg[dst[e]] += 1   (stored into the dinv buffer, converted in place)
// ---------------------------------------------------------------------------
__global__ __launch_bounds__(256) void deg_count(const int* __restrict__ dst,
                                                 float* __restrict__ deg, int nedges) {
  int e = blockIdx.x * blockDim.x + threadIdx.x;
  if (e < nedges) atomicAdd(&deg[dst[e]], 1.0f);
}

__global__ __launch_bounds__(256) void deg_to_dinv(float* __restrict__ deg, int nnodes) {
  int n = blockIdx.x * blockDim.x + threadIdx.x;
  if (n < nnodes) deg[n] = rsqrtf(deg[n] + 1.0f);   // +1 self loop
}

// ---------------------------------------------------------------------------
// Dense GEMM  C[N,128] = [relu](A[N,128]) @ W[128,128]  via
// V_WMMA_F32_16X16X4_F32.
//  - 8 waves/block: wave w owns a 16-column tile; preloads its 32 B fragments
//    (64 VGPRs) once and reuses them across ROWT row tiles.
//  - 80x128 A tile staged in LDS (stride 132 floats): fragment reads are
//    conflict-free ds_load_b64 (half-wave 0 -> banks {4r,4r+1},
//    half-wave 1 -> banks {4r+2,4r+3}).
// Fragment layouts (ISA 7.12.2):
//   A 16x4 f32: lane half h holds K={2h,2h+1}, row M=lane%16   (2 VGPRs)
//   B 4x16 f32: VGPR r, half h -> K=r+2h, N=lane%16            (2 VGPRs)
//   C 16x16 f32: VGPR r, half h -> M=r+8h, N=lane%16           (8 VGPRs)
// ---------------------------------------------------------------------------
__global__ __launch_bounds__(256) void gemm_wmma_f32(const float* __restrict__ A,
                                                     const float* __restrict__ W,
                                                     float* __restrict__ C,
                                                     int nrows, int relu_in) {
  __shared__ float at[TROWS * LSTR];
  const int tid  = threadIdx.x;
  const int lane = tid & 31;
  const int wave = tid >> 5;
  const int h = lane >> 4;         // lane half
  const int m = lane & 15;
  const int colBase = wave * 16;
  const int rowBase = blockIdx.x * TROWS;

  // --- preload this wave's B fragments for all 32 K-steps ---
  v2f bfrag[32];
#pragma unroll
  for (int kk = 0; kk < 32; ++kk) {
    const int k = kk * 4 + 2 * h;
    bfrag[kk].x = W[(size_t)(k + 0) * DIM + colBase + m];
    bfrag[kk].y = W[(size_t)(k + 1) * DIM + colBase + m];
  }

  // --- stage 80x128 A tile into LDS (optionally with fused ReLU) ---
  for (int i = tid; i < TROWS * 32; i += 256) {
    const int r = i >> 5;
    const int q = i & 31;
    int gr = rowBase + r;
    if (gr > nrows - 1) gr = nrows - 1;          // defensive clamp (N%80==0)
    float4 v = ((const float4*)(A + (size_t)gr * DIM))[q];
    if (relu_in) {
      v.x = fmaxf(v.x, 0.0f); v.y = fmaxf(v.y, 0.0f);
      v.z = fmaxf(v.z, 0.0f); v.w = fmaxf(v.w, 0.0f);
    }
    *(float4*)&at[r * LSTR + q * 4] = v;
  }
  __syncthreads();

  // --- 5 row tiles x 32 WMMA each, B held in registers ---
  for (int rt = 0; rt < ROWT; ++rt) {
    const float* __restrict__ arow = &at[(rt * 16 + m) * LSTR + 2 * h];
    v8f acc = {};
#pragma unroll
    for (int kk = 0; kk < 32; ++kk) {
      v2f a;
      a.x = arow[kk * 4 + 0];
      a.y = arow[kk * 4 + 1];
      acc = __builtin_amdgcn_wmma_f32_16x16x4_f32(
          /*neg_a=*/false, a, /*neg_b=*/false, bfrag[kk],
          /*c_mod=*/(short)0, acc, /*reuse_a=*/false, /*reuse_b=*/false);
    }
    float* __restrict__ crow = C + (size_t)(rowBase + rt * 16) * DIM + colBase;
#pragma unroll
    for (int r = 0; r < 8; ++r) crow[(size_t)(r + 8 * h) * DIM + m] = acc[r];
  }
}

// ---------------------------------------------------------------------------
// Init aggregation buffer with self-loop + bias: agg = h * dinv^2 + bias
// (replaces zero + post-combine passes)
// ---------------------------------------------------------------------------
__global__ __launch_bounds__(256) void init_agg(float* __restrict__ agg,
                                                const float* __restrict__ h,
                                                const float* __restrict__ dinv,
                                                const float* __restrict__ bias,
                                                int nnodes) {
  int t = blockIdx.x * blockDim.x + threadIdx.x;
  if (t >= nnodes * DIM) return;
  int n = t >> 7;
  int c = t & (DIM - 1);
  float di = dinv[n];
  agg[t] = h[t] * (di * di) + bias[c];
}

// ---------------------------------------------------------------------------
// Edge scatter: agg[dst] += h[src] * (dinv[src]*dinv[dst])
// 32 threads per edge, float4 per thread -> coalesced 512B gather per edge.
// ---------------------------------------------------------------------------
__global__ __launch_bounds__(256) void scatter_edges(const float* __restrict__ h,
                                                     const int* __restrict__ src,
                                                     const int* __restrict__ dst,
                                                     const float* __restrict__ dinv,
                                                     float* __restrict__ agg,
                                                     int nedges) {
  int t = blockIdx.x * blockDim.x + threadIdx.x;
  int e = t >> 5;
  if (e >= nedges) return;
  int q = t & 31;
  int s = src[e], d = dst[e];
  float w = dinv[s] * dinv[d];
  const float4* __restrict__ hv = (const float4*)(h + (size_t)s * DIM);
  float4 v = hv[q];
  float* __restrict__ out = agg + (size_t)d * DIM + q * 4;
  atomicAdd(out + 0, v.x * w);
  atomicAdd(out + 1, v.y * w);
  atomicAdd(out + 2, v.z * w);
  atomicAdd(out + 3, v.w * w);
}

// ---------------------------------------------------------------------------
// Global mean pool: psum[batch[n]] += h[n]; pcnt[batch[n]] += 1
// ---------------------------------------------------------------------------
__global__ __launch_bounds__(256) void pool_sum(const float* __restrict__ h,
                                                const int* __restrict__ batch,
                                                float* __restrict__ psum,
                                                float* __restrict__ pcnt,
                                                int nnodes) {
  int t = blockIdx.x * blockDim.x + threadIdx.x;
  if (t >= nnodes * DIM) return;
  int n = t >> 7;
  int c = t & (DIM - 1);
  int g = batch[n];
  atomicAdd(&psum[g * DIM + c], h[t]);
  if (c == 0) atomicAdd(&pcnt[g], 1.0f);
}

// ---------------------------------------------------------------------------
// Final head: out[g,c] = relu( (psum[g]/max(cnt,1)) @ Wlin + blin )
// ---------------------------------------------------------------------------
__global__ __launch_bounds__(256) void final_head(const float* __restrict__ psum,
                                                  const float* __restrict__ pcnt,
                                                  const float* __restrict__ Wlin,
                                                  const float* __restrict__ blin,
                                                  float* __restrict__ out) {
  int t = blockIdx.x * blockDim.x + threadIdx.x;
  if (t >= NGRAPH * DIM) return;
  int g = t >> 7;
  int c = t & (DIM - 1);
  float inv = 1.0f / fmaxf(pcnt[g], 1.0f);
  float acc = blin[c];
  const float* __restrict__ prow = psum + g * DIM;
#pragma unroll 4
  for (int k = 0; k < DIM; ++k) {
    acc = fmaf(prow[k] * inv, Wlin[k * DIM + c], acc);
  }
  out[t] = fmaxf(acc, 0.0f);
}

// ---------------------------------------------------------------------------
extern "C" void kernel_launch(void* const* d_in, const int* in_sizes, int n_in,
                              void* d_out, int out_size, void* d_ws, size_t ws_size,
                              hipStream_t stream) {
  const float* x    = (const float*)d_in[0];
  const int*   ei   = (const int*)  d_in[1];
  const int*   batch= (const int*)  d_in[2];
  const float* W1   = (const float*)d_in[3];
  const float* b1   = (const float*)d_in[4];
  const float* W2   = (const float*)d_in[5];
  const float* b2   = (const float*)d_in[6];
  const float* Wlin = (const float*)d_in[7];
  const float* blin = (const float*)d_in[8];
  float*       out  = (float*)d_out;

  const int N = in_sizes[0] / DIM;   // 50000
  const int E = in_sizes[1] / 2;     // 640000
  const int* src = ei;               // edge_index row 0
  const int* dst = ei + E;           // edge_index row 1
  const int ND = N * DIM;

  // carve scratch
  char* ws = (char*)d_ws;
  size_t off = 0;
  auto carve = [&](size_t bytes) -> float* {
    float* p = (float*)(ws + off);
    off = (off + bytes + 255) & ~(size_t)255;
    return p;
  };
  float* dinv = carve((size_t)N * 4);             // degree then rsqrt, in place
  float* bufA = carve((size_t)ND * 4);            // h = X@W
  float* bufB = carve((size_t)ND * 4);            // agg / layer output
  float* psum = carve((size_t)NGRAPH * DIM * 4);
  float* pcnt = carve((size_t)NGRAPH * 4);
  (void)ws_size; (void)n_in; (void)out_size;

  const dim3 blk(256);
  const int gemmBlocks  = (N + TROWS - 1) / TROWS;   // 625 (N % 80 == 0)
  const int edgeThreads = E * 32;

  // --- degrees -> dinv ---
  zero_f32   <<<(N + 255) / 256, blk, 0, stream>>>(dinv, N);
  deg_count  <<<(E + 255) / 256, blk, 0, stream>>>(dst, dinv, E);
  deg_to_dinv<<<(N + 255) / 256, blk, 0, stream>>>(dinv, N);

  // --- layer 1:  h1 = x@W1 ; agg1 = h1*dinv^2 + b1 + scatter ---
  gemm_wmma_f32<<<gemmBlocks, blk, 0, stream>>>(x, W1, bufA, N, /*relu_in=*/0);
  init_agg     <<<(ND + 255) / 256, blk, 0, stream>>>(bufB, bufA, dinv, b1, N);
  scatter_edges<<<(edgeThreads + 255) / 256, blk, 0, stream>>>(bufA, src, dst, dinv, bufB, E);
  // bufB now holds pre-ReLU layer-1 output; ReLU is fused into GEMM-2's A load

  // --- layer 2:  h2 = relu(out1)@W2 ; agg2 = h2*dinv^2 + b2 + scatter ---
  gemm_wmma_f32<<<gemmBlocks, blk, 0, stream>>>(bufB, W2, bufA, N, /*relu_in=*/1);
  init_agg     <<<(ND + 255) / 256, blk, 0, stream>>>(bufB, bufA, dinv, b2, N);
  scatter_edges<<<(edgeThreads + 255) / 256, blk, 0, stream>>>(bufA, src, dst, dinv, bufB, E);

  // --- mean pool + linear head ---
  zero_f32<<<(NGRAPH * DIM + 255) / 256, blk, 0, stream>>>(psum, NGRAPH * DIM);
  zero_f32<<<1, blk, 0, stream>>>(pcnt, NGRAPH);
  pool_sum<<<(ND + 255) / 256, blk, 0, stream>>>(bufB, batch, psum, pcnt, N);
  final_head<<<(NGRAPH * DIM + 255) / 256, blk, 0, stream>>>(psum, pcnt, Wlin, blin, out);
}